// HashEncoding_58823872086056
// MI455X (gfx1250) — compile-verified
//
#include <hip/hip_runtime.h>
#include <stdint.h>

// ---- problem constants (match reference) ----
#define NUM_LEVELS   16
#define LOG2_HASH    19
#define HASH_SIZE    (1u << LOG2_HASH)
#define HASH_MASK    (HASH_SIZE - 1u)
#define FEATS        2
#define PRIME1       2654435761u
#define PRIME2       805459861u

#define POINTS_PER_BLOCK 64          // 256 threads, 4 threads (level-quads) per point
#define THREADS_PER_BLOCK 256

// native clang vector type (HIP's float4 is a class -> rejected by nontemporal builtins)
typedef float v4f __attribute__((ext_vector_type(4)));
typedef float v2f __attribute__((ext_vector_type(2)));

// int(16 * 32**(i/15)) for i in 0..15 (exact values of the reference's RESOLUTIONS)
__device__ __constant__ float c_res[NUM_LEVELS] = {
    16.f, 20.f, 25.f, 32.f, 40.f, 50.f, 64.f, 80.f,
    101.f, 128.f, 161.f, 203.f, 256.f, 322.f, 406.f, 512.f
};

__global__ __launch_bounds__(THREADS_PER_BLOCK)
void hash_encoding_kernel(const float* __restrict__ coords,   // [N,3]
                          const float* __restrict__ tables,   // [16, 2^19, 2]
                          float* __restrict__ out,            // [N,32]
                          int n_points)
{
    // ---- stage this block's coords (64 points * 12B = 768B = 96 qwords) into LDS
    //      via the CDNA5 async global->LDS path (ASYNCcnt) ----
    __shared__ float s_coords[POINTS_PER_BLOCK * 3];

    const int tib      = threadIdx.x;
    const int block_p0 = blockIdx.x * POINTS_PER_BLOCK;
    const int points_here = min(POINTS_PER_BLOCK, n_points - block_p0);

    // lanes 0..95 == waves 0..2 entirely (EXEC all-ones per issuing wave)
    if (tib < 96 && (tib * 8) < (points_here * 12)) {
        // low 32 bits of a generic shared address are the raw LDS byte offset
        uint32_t lds_off = (uint32_t)(uintptr_t)(&s_coords[0]) + (uint32_t)(tib * 8);
        uint64_t gaddr   = (uint64_t)(uintptr_t)coords
                         + (uint64_t)block_p0 * 12ull + (uint64_t)(tib * 8);
        asm volatile("global_load_async_to_lds_b64 %0, %1, off"
                     :: "v"(lds_off), "v"(gaddr)
                     : "memory");
    }
    // wait for our wave's async copies (no-op for waves that issued none)
    asm volatile("s_wait_asynccnt 0" ::: "memory");
    __syncthreads();

    const int lq    = tib & 3;          // which quad of levels: levels 4*lq .. 4*lq+3
    const int lp    = tib >> 2;         // local point index 0..63
    const int point = block_p0 + lp;
    if (point >= n_points) return;

    const float cx = s_coords[lp * 3 + 0];
    const float cy = s_coords[lp * 3 + 1];
    const float cz = s_coords[lp * 3 + 2];

    const int level0 = lq * 4;

    float vals[8];
#pragma unroll
    for (int j = 0; j < 4; ++j) {
        const int level = level0 + j;
        const float r = c_res[level];

        // reference: grid = floor(c * r) as uint32; hash in uint32 == int64 mod 2^32
        const uint32_t g0 = (uint32_t)floorf(cx * r);
        const uint32_t g1 = (uint32_t)floorf(cy * r);
        const uint32_t g2 = (uint32_t)floorf(cz * r);
        const uint32_t h  = g0 ^ (g1 * PRIME1) ^ (g2 * PRIME2);
        const uint32_t idx = h & HASH_MASK;

        // tables are 64MB total -> resident in the 192MB L2; default (RT) loads
        const v2f* tp =
            (const v2f*)(tables + (size_t)level * (size_t)(HASH_SIZE * FEATS));
        const v2f f = tp[idx];
        vals[j * 2 + 0] = f.x;
        vals[j * 2 + 1] = f.y;
    }

    // 8 contiguous floats per thread; wave covers a dense contiguous 1KB span.
    // Non-temporal: stream 256MB of output past L2 so the tables stay resident.
    v4f a = { vals[0], vals[1], vals[2], vals[3] };
    v4f b = { vals[4], vals[5], vals[6], vals[7] };
    v4f* op = (v4f*)(out + (size_t)point * (NUM_LEVELS * FEATS) + (size_t)level0 * 2);
    __builtin_nontemporal_store(a, op);
    __builtin_nontemporal_store(b, op + 1);
}

extern "C" void kernel_launch(void* const* d_in, const int* in_sizes, int n_in,
                              void* d_out, int out_size, void* d_ws, size_t ws_size,
                              hipStream_t stream)
{
    const float* coords = (const float*)d_in[0];   // [N,3] float32
    const float* tables = (const float*)d_in[1];   // [16, 2^19, 2] float32
    float* out = (float*)d_out;                    // [N,32] float32

    const int n_points = in_sizes[0] / 3;
    const int blocks   = (n_points + POINTS_PER_BLOCK - 1) / POINTS_PER_BLOCK;

    hipLaunchKernelGGL(hash_encoding_kernel,
                       dim3(blocks), dim3(THREADS_PER_BLOCK), 0, stream,
                       coords, tables, out, n_points);
}